// NeuSSampler_30468497998319
// MI455X (gfx1250) — compile-verified
//
#include <hip/hip_runtime.h>
#include <math.h>

// NeuS hierarchical sampler for MI455X (gfx1250, wave32).
// One ray per lane; per-ray arrays in conflict-free lane-interleaved LDS.
// Inputs staged via GLOBAL_LOAD_ASYNC_TO_LDS (ASYNCcnt path).

#define NSAMP 64          // NUM_SAMPLES
#define NUP   4           // NUM_UPSAMPLE
#define NPER  16          // NUM_IMPORTANCE / NUM_UPSAMPLE
#define NBQ   (NPER + 1)  // samples drawn per round
#define MAXB  129         // final bins per ray (65 + 4*16)
#define BLK   32          // one wave32 per block

__device__ __forceinline__ float fast_rcp(float x) {
#if __has_builtin(__builtin_amdgcn_rcpf)
    return __builtin_amdgcn_rcpf(x);
#else
    return 1.0f / x;
#endif
}

__device__ __forceinline__ float sigmoid_fast(float x) {
    // 1/(1+exp(-x)) via v_exp_f32 + v_rcp_f32
    return fast_rcp(1.0f + __expf(-x));
}

__global__ __launch_bounds__(BLK)
void neus_sampler_kernel(const float* __restrict__ origins,
                         const float* __restrict__ directions,
                         const float* __restrict__ nears,
                         const float* __restrict__ fars,
                         float* __restrict__ out,
                         int R_)
{
    // Regions (dwords): bins 129*32, weights 128*32, new_bins 17*32 = 274*32.
    // The weights+new_bins region (>=4640 dw) is reused as the 32x133 transpose
    // buffer (4256 dw) for the coalesced output stage.
    __shared__ float sm[(MAXB + (MAXB - 1) + NBQ) * BLK];
    // Input staging: origins[96] | directions[96] | nears[32] | fars[32]
    __shared__ float inbuf[256];

    const int tid = threadIdx.x;
    const int rb  = blockIdx.x * BLK;
    const int ray = rb + tid;
    const bool active = (ray < R_);

    // ---- stage this block's inputs into LDS via async global->LDS DMA ----
    if (rb + BLK <= R_) {
        const unsigned lb = (unsigned)(uintptr_t)(&inbuf[0]); // wave-relative DS offset
        const float* og = origins    + (long long)3 * rb;
        const float* dg = directions + (long long)3 * rb;
        #pragma unroll
        for (int s = 0; s < 3; ++s) {
            unsigned d0 = lb + 4u * (unsigned)(tid + 32 * s);
            unsigned long long g0 = (unsigned long long)(uintptr_t)(og + tid + 32 * s);
            asm volatile("global_load_async_to_lds_b32 %0, %1, off"
                         :: "v"(d0), "v"(g0) : "memory");
            unsigned d1 = lb + 4u * (unsigned)(96 + tid + 32 * s);
            unsigned long long g1 = (unsigned long long)(uintptr_t)(dg + tid + 32 * s);
            asm volatile("global_load_async_to_lds_b32 %0, %1, off"
                         :: "v"(d1), "v"(g1) : "memory");
        }
        {
            unsigned dn = lb + 4u * (unsigned)(192 + tid);
            unsigned long long gn = (unsigned long long)(uintptr_t)(nears + rb + tid);
            asm volatile("global_load_async_to_lds_b32 %0, %1, off"
                         :: "v"(dn), "v"(gn) : "memory");
            unsigned df = lb + 4u * (unsigned)(224 + tid);
            unsigned long long gf = (unsigned long long)(uintptr_t)(fars + rb + tid);
            asm volatile("global_load_async_to_lds_b32 %0, %1, off"
                         :: "v"(df), "v"(gf) : "memory");
        }
        asm volatile("s_wait_asynccnt 0x0" ::: "memory");
    } else if (active) {
        // tail block fallback: guarded direct loads
        __builtin_prefetch(&origins[3 * ray], 0, 0);
        inbuf[3 * tid + 0]  = origins[3 * ray + 0];
        inbuf[3 * tid + 1]  = origins[3 * ray + 1];
        inbuf[3 * tid + 2]  = origins[3 * ray + 2];
        inbuf[96 + 3 * tid + 0] = directions[3 * ray + 0];
        inbuf[96 + 3 * tid + 1] = directions[3 * ray + 1];
        inbuf[96 + 3 * tid + 2] = directions[3 * ray + 2];
        inbuf[192 + tid] = nears[ray];
        inbuf[224 + tid] = fars[ray];
    }
    __syncthreads();

    float* binsb = &sm[0];
    float* wb    = &sm[MAXB * BLK];
    float* nbuf  = &sm[(MAXB + MAXB - 1) * BLK];
#define BINS(i) binsb[(i) * BLK + tid]
#define WB(i)   wb[(i) * BLK + tid]
#define NEWB(i) nbuf[(i) * BLK + tid]

    float nr = 0.f, fr = 1.f;

    if (active) {
        float ox = inbuf[3 * tid + 0], oy = inbuf[3 * tid + 1], oz = inbuf[3 * tid + 2];
        float dx = inbuf[96 + 3 * tid + 0], dy = inbuf[96 + 3 * tid + 1], dz = inbuf[96 + 3 * tid + 2];
        nr = inbuf[192 + tid]; fr = inbuf[224 + tid];
        float dn = 1.0f / sqrtf(dx * dx + dy * dy + dz * dz);
        dx *= dn; dy *= dn; dz *= dn;

        // bins = linspace(0,1,65)
        #pragma unroll
        for (int k = 0; k <= NSAMP; ++k) BINS(k) = (float)k * (1.0f / 64.0f);

        int nbins = NSAMP + 1;

        for (int it = 0; it < NUP; ++it) {
            const float inv_s = 64.0f * (float)(1 << it);
            const int M = nbins - 1;            // #zs samples == #weights (after pad)

            // ---- pass 1: streaming sdf -> alphas -> weights (store w+HIST_PAD) ----
            float b0v      = BINS(0);
            float z_prev   = fr * b0v + nr * (1.0f - b0v);
            float px = ox + dx * z_prev, py = oy + dy * z_prev, pz = oz + dz * z_prev;
            float sdf_prev = sqrtf(px * px + py * py + pz * pz) - 1.0f;
            float rawcos_prev = 0.0f;           // prev_cos = concat(0, cos_val[:-1]) (unclipped)
            float trans = 1.0f;
            float wsum  = 0.0f;

            #pragma unroll 2
            for (int k = 1; k < M; ++k) {
                float bv  = BINS(k);
                float z   = fr * bv + nr * (1.0f - bv);
                float qx = ox + dx * z, qy = oy + dy * z, qz = oz + dz * z;
                float sdf = sqrtf(qx * qx + qy * qy + qz * qz) - 1.0f;

                float delta  = z - z_prev;
                float mid    = 0.5f * (sdf_prev + sdf);
                float rawcos = (sdf - sdf_prev) * fast_rcp(delta + 1e-5f);
                float cv     = fminf(rawcos_prev, rawcos);
                cv           = fminf(fmaxf(cv, -1000.0f), 0.0f);

                float pe = mid - cv * delta * 0.5f;
                float ne = mid + cv * delta * 0.5f;
                float pc = sigmoid_fast(pe * inv_s);
                float nc = sigmoid_fast(ne * inv_s);
                float a  = (pc - nc + 1e-5f) * fast_rcp(pc + 1e-5f);

                float w  = a * trans + 1e-5f;   // weights + HIST_PAD
                trans   *= (1.0f - a + 1e-7f);
                WB(k - 1) = w;
                wsum     += w;

                rawcos_prev = rawcos;
                z_prev = z; sdf_prev = sdf;
            }
            WB(M - 1) = 1e-5f;                  // trailing zero weight + HIST_PAD
            wsum     += 1e-5f;

            float pad  = fmaxf(1e-5f - wsum, 0.0f);   // relu(EPS - wsum)
            float addw = pad / (float)M;
            wsum      += pad;
            float invw = 1.0f / wsum;

            // ---- pass 2: forward inverse-CDF sweep (== searchsorted side='right') ----
            float S = 0.0f, cprev = 0.0f;
            int   j = 0;
            float bprev = BINS(0);
            for (int k = 1; k <= M; ++k) {
                S += (WB(k - 1) + addw) * invw;
                float ccur = fminf(1.0f, S);
                float bcur = BINS(k);
                while (j < NBQ) {
                    float u = (float)(2 * j + 1) * (1.0f / 34.0f);  // stratified midpoints
                    if (!(u < ccur)) break;                          // cdf[k-1] <= u < cdf[k]
                    float t = (u - cprev) / (ccur - cprev);
                    t = fminf(fmaxf(t, 0.0f), 1.0f);
                    NEWB(j) = bprev + t * (bcur - bprev);
                    ++j;
                }
                cprev = ccur;
                bprev = bcur;
            }
            float blast = BINS(M);
            while (j < NBQ) { NEWB(j) = blast; ++j; }  // u >= cdf[M] fallback -> bins[M]

            // ---- in-place backward merge of bins[:-1] (M) with new_bins[:-1] (16) ----
            float endv = fmaxf(blast, NEWB(NPER));
            int outI = M + NPER;               // new last index
            BINS(outI) = endv;
            --outI;
            int p = M - 1, q = NPER - 1;
            while (outI >= 0) {
                float v;
                if (q < 0)      { v = BINS(p); --p; }
                else if (p < 0) { v = NEWB(q); --q; }
                else {
                    float bv = BINS(p), nv = NEWB(q);
                    if (bv >= nv) { v = bv; --p; } else { v = nv; --q; }
                }
                BINS(outI) = v;
                --outI;
            }
            nbins += NPER;
        }

        // euclidean conversion into transposed LDS (stride 133: coprime w/ 64 banks)
        float* tb = &sm[MAXB * BLK];
        #pragma unroll 4
        for (int k = 0; k < MAXB; ++k) {
            float b = BINS(k);
            tb[tid * 133 + k] = fr * b + nr * (1.0f - b);
        }
    }

    __syncthreads();

    // ---- coalesced output: wave writes 32 rows cooperatively ----
    {
        const float* tb = &sm[MAXB * BLK];
        const long long base = (long long)blockIdx.x * BLK;
        int lr = tid / MAXB;     // == 0
        int k  = tid % MAXB;     // == tid
        for (int e = tid; e < BLK * MAXB; e += BLK) {
            if (base + lr < R_) {
                out[base * MAXB + e] = tb[lr * 133 + k];
            }
            k += BLK;
            if (k >= MAXB) { k -= MAXB; ++lr; }
        }
    }
#undef BINS
#undef WB
#undef NEWB
}

extern "C" void kernel_launch(void* const* d_in, const int* in_sizes, int n_in,
                              void* d_out, int out_size, void* d_ws, size_t ws_size,
                              hipStream_t stream) {
    const float* origins    = (const float*)d_in[0];
    const float* directions = (const float*)d_in[1];
    const float* nears      = (const float*)d_in[2];
    const float* fars       = (const float*)d_in[3];
    float* out = (float*)d_out;

    int R_ = in_sizes[0] / 3;
    int grid = (R_ + BLK - 1) / BLK;
    hipLaunchKernelGGL(neus_sampler_kernel, dim3(grid), dim3(BLK), 0, stream,
                       origins, directions, nears, fars, out, R_);
}